// MyEmbedding_4191888081309
// MI455X (gfx1250) — compile-verified
//
#include <hip/hip_runtime.h>
#include <cstdint>

// Embedding gather: y[t, e] = W[e, x[t]] + b[e]
//   x: [64*2048] int32, W: [128, 50257] f32 row-major, b: [128] f32
//   out: [64*2048, 128] f32
// Bandwidth-bound: ~90 MB of HBM traffic -> ~4 us at 23.3 TB/s. No WMMA
// (one-hot GEMM would be ~4 orders of magnitude more FLOPs than the gather).
// CDNA5 paths used: TDM tensor_load_to_lds to stage bias into LDS,
// s_wait_tensorcnt, global_prefetch_b8, wave32-shaped 512B coalesced stores.

#define EMB   128
#define VOCABSZ 50257

typedef __attribute__((ext_vector_type(4))) unsigned int u32x4;
typedef __attribute__((ext_vector_type(8))) int          i32x8;
typedef __attribute__((ext_vector_type(4))) int          i32x4;

__global__ __launch_bounds__(256) void emb_gather_kernel(
    const int* __restrict__ idx, const float* __restrict__ W,
    const float* __restrict__ bias, float* __restrict__ out, int ntok)
{
    __shared__ float lds_b[EMB];

#if __has_builtin(__builtin_amdgcn_tensor_load_to_lds)
    {
        // --- TDM descriptor: 1-D tile of 128 x 4B = the bias vector ---
        unsigned long long ga = (unsigned long long)(uintptr_t)bias;
        unsigned int lds_off  = (unsigned int)(uintptr_t)&lds_b[0]; // low 32b of flat addr = LDS offset
        // Group 0: count=1 | lds_addr | global_addr[56:0] | type=2 (bits 127:126)
        u32x4 g0 = { 1u,
                     lds_off,
                     (unsigned int)(ga & 0xFFFFFFFFu),
                     (unsigned int)((ga >> 32) & 0x01FFFFFFu) | 0x80000000u };
        // Group 1: data_size=2 (4B) @ [17:16]; tensor_dim0=128; tensor_dim1=1;
        //          tile_dim0=128; tile_dim1=1; dim0_stride=128; dim1_stride=128
        i32x8 g1 = { 0x00020000,        // wg_mask=0, data_size=2
                     (EMB << 16),       // tensor_dim0[15:0] in [31:16]
                     (1   << 16),       // tensor_dim1[15:0] in [31:16]
                     (EMB << 16),       // tile_dim0 in [31:16]
                     1,                 // tile_dim1 = 1
                     EMB,               // tensor_dim0_stride low32
                     (EMB << 16),       // tensor_dim1_stride[15:0] in [31:16]
                     0 };
        i32x4 gz4 = { 0, 0, 0, 0 };
#if __has_include(<hip/amd_detail/amd_gfx1250_TDM.h>)
        i32x8 gz8 = { 0, 0, 0, 0, 0, 0, 0, 0 };
        __builtin_amdgcn_tensor_load_to_lds(g0, g1, gz4, gz4, gz8, 0); // 6-arg toolchain
#else
        __builtin_amdgcn_tensor_load_to_lds(g0, g1, gz4, gz4, 0);      // 5-arg ROCm 7.2
#endif
        __builtin_amdgcn_s_wait_tensorcnt(0); // each wave waits on its own copy
    }
#else
    if (threadIdx.x < EMB) lds_b[threadIdx.x] = bias[threadIdx.x];
#endif
    __syncthreads();

    // wave32 tiling: 32 lanes x float4 = one full 128-wide embedding row
    const int lane = threadIdx.x & 31;
    const int e0   = lane << 2;
    const float4 bv = *reinterpret_cast<const float4*>(&lds_b[e0]); // ds_load_b128

    int t = blockIdx.x * (blockDim.x >> 5) + (threadIdx.x >> 5);
    const int stride = gridDim.x * (blockDim.x >> 5);

    for (; t < ntok; t += stride) {
        __builtin_prefetch(idx + t + stride, 0, 0);   // global_prefetch_b8 (speculative)
        const int id = idx[t];
        const float* col = W + id;                    // column id, rows strided by VOCAB
        float4 v;
        v.x = col[(size_t)(e0 + 0) * VOCABSZ] + bv.x;
        v.y = col[(size_t)(e0 + 1) * VOCABSZ] + bv.y;
        v.z = col[(size_t)(e0 + 2) * VOCABSZ] + bv.z;
        v.w = col[(size_t)(e0 + 3) * VOCABSZ] + bv.w;
        // wave writes one contiguous 512B burst per token
        *reinterpret_cast<float4*>(out + (size_t)t * EMB + e0) = v;   // global_store_b128
    }
}

extern "C" void kernel_launch(void* const* d_in, const int* in_sizes, int n_in,
                              void* d_out, int out_size, void* d_ws, size_t ws_size,
                              hipStream_t stream) {
    const int*   idx = (const int*)d_in[0];    // x: [64, 2048] int32
    const float* W   = (const float*)d_in[1];  // [128, 50257] f32
    const float* b   = (const float*)d_in[2];  // [128] f32
    float*       out = (float*)d_out;          // [64, 2048, 128] f32

    const int ntok = in_sizes[0];              // 131072 tokens
    const int tokPerBlock = 256 / 32;          // 8 waves -> 8 tokens per pass
    int blocks = (ntok + tokPerBlock - 1) / tokPerBlock;
    if (blocks > 2048) blocks = 2048;          // grid-stride; amortize bias staging
    if (blocks < 1) blocks = 1;

    emb_gather_kernel<<<blocks, 256, 0, stream>>>(idx, W, b, out, ntok);
}